// SimpleGraphEncoder_22325240004854
// MI455X (gfx1250) — compile-verified
//
#include <hip/hip_runtime.h>
#include <hip/hip_bf16.h>

typedef __attribute__((ext_vector_type(2))) float v2f;
typedef __attribute__((ext_vector_type(8))) float v8f;

#define N_NODES 100000
#define N_EDGES 1600000
#define FEAT    128
#define HID     64
#define LAYERS  3
#define MTILES  (N_NODES / 16)   // 6250

// ---------- helpers: order-preserving float<->uint key for atomic max ----------
__device__ __forceinline__ unsigned f2key(float f) {
    unsigned u = __float_as_uint(f);
    return (u & 0x80000000u) ? ~u : (u | 0x80000000u);
}
__device__ __forceinline__ float key2f(unsigned k) {
    return (k & 0x80000000u) ? __uint_as_float(k & 0x7FFFFFFFu) : __uint_as_float(~k);
}

// ---------------------------------------------------------------------------
// Wave-level fp32 WMMA GEMM stripe: one wave computes a 16(M) x 64(N) tile of
// OUT = A[M_TOTAL,K] @ B[K,64], using 4 independent accumulator chains so the
// four v_wmma_f32_16x16x4_f32 per K-step have no RAW dependency on each other.
// A is loaded once per K-step (b64), B (<=32KB) stays L2-resident.
// ---------------------------------------------------------------------------
template <int K>
__device__ __forceinline__ void wmma_stripe(const float* __restrict__ A,
                                            const float* __restrict__ B,
                                            int mtile, int lane,
                                            v8f& c0, v8f& c1, v8f& c2, v8f& c3) {
    const int half = lane >> 4;        // 0: lanes 0-15, 1: lanes 16-31
    const int l    = lane & 15;
    const int row  = mtile * 16 + l;   // A row (M)
    const int kb   = half * 2;         // K sub-offset per ISA 16x4 layout

    for (int kk = 0; kk < K; kk += 4) {
        v2f a, b0, b1, b2, b3;
        a.x  = A[row * K + kk + kb];
        a.y  = A[row * K + kk + kb + 1];
        const float* Brow0 = B + (kk + kb) * HID;
        const float* Brow1 = Brow0 + HID;
        b0.x = Brow0[l];       b0.y = Brow1[l];
        b1.x = Brow0[16 + l];  b1.y = Brow1[16 + l];
        b2.x = Brow0[32 + l];  b2.y = Brow1[32 + l];
        b3.x = Brow0[48 + l];  b3.y = Brow1[48 + l];
        c0 = __builtin_amdgcn_wmma_f32_16x16x4_f32(false, a, false, b0, (short)0, c0, false, false);
        c1 = __builtin_amdgcn_wmma_f32_16x16x4_f32(false, a, false, b1, (short)0, c1, false, false);
        c2 = __builtin_amdgcn_wmma_f32_16x16x4_f32(false, a, false, b2, (short)0, c2, false, false);
        c3 = __builtin_amdgcn_wmma_f32_16x16x4_f32(false, a, false, b3, (short)0, c3, false, false);
    }
}

// h = X[100000,128] @ W[128,64] + b
__global__ __launch_bounds__(256) void gemm_proj(const float* __restrict__ X,
                                                 const float* __restrict__ W,
                                                 const float* __restrict__ bias,
                                                 float* __restrict__ H) {
    const int lane  = threadIdx.x & 31;
    const int wave  = threadIdx.x >> 5;
    const int mtile = blockIdx.x * 8 + wave;
    if (mtile >= MTILES) return;

    v8f c0 = {}, c1 = {}, c2 = {}, c3 = {};
    wmma_stripe<FEAT>(X, W, mtile, lane, c0, c1, c2, c3);

    const int half  = lane >> 4;
    const int l     = lane & 15;
    const int mbase = mtile * 16 + half * 8;
#pragma unroll
    for (int r = 0; r < 8; ++r) {
        float* out = H + (size_t)(mbase + r) * HID;
        out[l]      = c0[r] + bias[l];
        out[16 + l] = c1[r] + bias[16 + l];
        out[32 + l] = c2[r] + bias[32 + l];
        out[48 + l] = c3[r] + bias[48 + l];
    }
}

// h = h + relu(MSG[100000,64] @ Wc[64,64] + bc)
__global__ __launch_bounds__(256) void gemm_conv_relu_add(const float* __restrict__ MSG,
                                                          const float* __restrict__ Wc,
                                                          const float* __restrict__ bc,
                                                          float* __restrict__ H) {
    const int lane  = threadIdx.x & 31;
    const int wave  = threadIdx.x >> 5;
    const int mtile = blockIdx.x * 8 + wave;
    if (mtile >= MTILES) return;

    v8f c0 = {}, c1 = {}, c2 = {}, c3 = {};
    wmma_stripe<HID>(MSG, Wc, mtile, lane, c0, c1, c2, c3);

    const int half  = lane >> 4;
    const int l     = lane & 15;
    const int mbase = mtile * 16 + half * 8;
#pragma unroll
    for (int r = 0; r < 8; ++r) {
        float* out = H + (size_t)(mbase + r) * HID;
        out[l]      += fmaxf(c0[r] + bc[l],      0.0f);
        out[16 + l] += fmaxf(c1[r] + bc[16 + l], 0.0f);
        out[32 + l] += fmaxf(c2[r] + bc[32 + l], 0.0f);
        out[48 + l] += fmaxf(c3[r] + bc[48 + l], 0.0f);
    }
}

// ---------------------------------------------------------------------------
// a_src[i] = h[i].wa[0:64],  a_dst[i] = h[i].wa[64:128]
// ---------------------------------------------------------------------------
__global__ __launch_bounds__(256) void attn_matvec(const float* __restrict__ H,
                                                   const float* __restrict__ wa,
                                                   float* __restrict__ a_src,
                                                   float* __restrict__ a_dst) {
    __shared__ float sw[2 * HID];
    if (threadIdx.x < 2 * HID) sw[threadIdx.x] = wa[threadIdx.x];
    __syncthreads();
    int i = blockIdx.x * blockDim.x + threadIdx.x;
    if (i >= N_NODES) return;
    float s = 0.f, d = 0.f;
#pragma unroll 8
    for (int j = 0; j < HID; ++j) {
        float hv = H[(size_t)i * HID + j];
        s += hv * sw[j];
        d += hv * sw[HID + j];
    }
    a_src[i] = s;
    a_dst[i] = d;
}

// ---------------------------------------------------------------------------
// msg = 0; red[0] = -inf key; red[1] = 0
// ---------------------------------------------------------------------------
__global__ __launch_bounds__(256) void zero_msg_init(float* __restrict__ msg,
                                                     float* __restrict__ red) {
    int i = blockIdx.x * blockDim.x + threadIdx.x;
    if (i < N_NODES * HID) msg[i] = 0.0f;
    if (i == 0) {
        ((unsigned*)red)[0] = 0u;   // key(-inf)
        red[1] = 0.0f;              // softmax denominator
    }
}

// ---------------------------------------------------------------------------
// e = leaky_relu(a_src[src]+a_dst[dst]+b, 0.2); global max -> red[0] (as key)
// ---------------------------------------------------------------------------
__global__ __launch_bounds__(256) void edge_e_max(const int* __restrict__ src,
                                                  const int* __restrict__ dst,
                                                  const float* __restrict__ a_src,
                                                  const float* __restrict__ a_dst,
                                                  const float* __restrict__ b_attn,
                                                  int layer,
                                                  float* __restrict__ e,
                                                  unsigned* __restrict__ redmax) {
    __shared__ float sm[256];
    int t = blockIdx.x * 256 + threadIdx.x;
    float v = -3.402823466e38f;
    if (t < N_EDGES) {
        float x = a_src[src[t]] + a_dst[dst[t]] + b_attn[layer];
        x = (x >= 0.0f) ? x : 0.2f * x;
        e[t] = x;
        v = x;
    }
    sm[threadIdx.x] = v;
    __syncthreads();
    for (int s = 128; s > 0; s >>= 1) {
        if (threadIdx.x < s) sm[threadIdx.x] = fmaxf(sm[threadIdx.x], sm[threadIdx.x + s]);
        __syncthreads();
    }
    if (threadIdx.x == 0) atomicMax(redmax, f2key(sm[0]));
}

// ---------------------------------------------------------------------------
// e = exp(e - max); global sum -> red[1]
// ---------------------------------------------------------------------------
__global__ __launch_bounds__(256) void edge_exp_sum(float* __restrict__ e,
                                                    float* __restrict__ red) {
    __shared__ float sm[256];
    const float m = key2f(((const unsigned*)red)[0]);
    int t = blockIdx.x * 256 + threadIdx.x;
    float v = 0.0f;
    if (t < N_EDGES) {
        float w = __expf(e[t] - m);
        e[t] = w;
        v = w;
    }
    sm[threadIdx.x] = v;
    __syncthreads();
    for (int s = 128; s > 0; s >>= 1) {
        if (threadIdx.x < s) sm[threadIdx.x] += sm[threadIdx.x + s];
        __syncthreads();
    }
    if (threadIdx.x == 0) atomicAdd(&red[1], sm[0]);
}

// ---------------------------------------------------------------------------
// msg[dst] += (e/sum) * h[src]   — 4 features per thread, float4 gather
// ---------------------------------------------------------------------------
__global__ __launch_bounds__(256) void scatter_msg(const int* __restrict__ src,
                                                   const int* __restrict__ dst,
                                                   const float* __restrict__ e,
                                                   const float* __restrict__ red,
                                                   const float* __restrict__ H,
                                                   float* __restrict__ msg) {
    int idx = blockIdx.x * 256 + threadIdx.x;   // E*16 = 25.6M < 2^31
    int edge = idx >> 4;
    if (edge >= N_EDGES) return;
    int fq = (idx & 15) * 4;
    const float inv = 1.0f / red[1];
    const float a = e[edge] * inv;
    const int s = src[edge];
    const int d = dst[edge];
    const float4 hv = *(const float4*)(H + (size_t)s * HID + fq);
    float* mp = msg + (size_t)d * HID + fq;
    atomicAdd(mp + 0, a * hv.x);
    atomicAdd(mp + 1, a * hv.y);
    atomicAdd(mp + 2, a * hv.z);
    atomicAdd(mp + 3, a * hv.w);
}

// ---------------------------------------------------------------------------
extern "C" void kernel_launch(void* const* d_in, const int* in_sizes, int n_in,
                              void* d_out, int out_size, void* d_ws, size_t ws_size,
                              hipStream_t stream) {
    const float* X  = (const float*)d_in[0];   // [N, FEAT]
    const int*   EI = (const int*)d_in[1];     // [2, E]
    const float* Wp = (const float*)d_in[2];   // [FEAT, HID]
    const float* bp = (const float*)d_in[3];   // [HID]
    const float* Wa = (const float*)d_in[4];   // [LAYERS, 2*HID]
    const float* ba = (const float*)d_in[5];   // [LAYERS]
    const float* Wc = (const float*)d_in[6];   // [LAYERS, HID, HID]
    const float* bc = (const float*)d_in[7];   // [LAYERS, HID]

    float* H   = (float*)d_out;                    // [N, HID] — running h
    float* ws  = (float*)d_ws;
    float* msg = ws;                               // N*HID
    float* a_s = msg + (size_t)N_NODES * HID;      // N
    float* a_d = a_s + N_NODES;                    // N
    float* e   = a_d + N_NODES;                    // E
    float* red = e + N_EDGES;                      // [0]=max key (uint), [1]=sum

    const int* src = EI;
    const int* dst = EI + N_EDGES;

    const int nodeBlocks = (N_NODES + 255) / 256;
    const int msgBlocks  = (N_NODES * HID + 255) / 256;
    const int edgeBlocks = (N_EDGES + 255) / 256;
    const int scatBlocks = (N_EDGES * 16 + 255) / 256;
    const int gemmBlocks = (MTILES + 7) / 8;       // 8 waves/block, 1 stripe/wave

    gemm_proj<<<gemmBlocks, 256, 0, stream>>>(X, Wp, bp, H);

    for (int L = 0; L < LAYERS; ++L) {
        attn_matvec<<<nodeBlocks, 256, 0, stream>>>(H, Wa + L * 2 * HID, a_s, a_d);
        zero_msg_init<<<msgBlocks, 256, 0, stream>>>(msg, red);
        edge_e_max<<<edgeBlocks, 256, 0, stream>>>(src, dst, a_s, a_d, ba, L, e,
                                                   (unsigned*)red);
        edge_exp_sum<<<edgeBlocks, 256, 0, stream>>>(e, red);
        scatter_msg<<<scatBlocks, 256, 0, stream>>>(src, dst, e, red, H, msg);
        gemm_conv_relu_add<<<gemmBlocks, 256, 0, stream>>>(msg, Wc + L * HID * HID,
                                                           bc + L * HID, H);
    }
}